// LDPCBeliefPropagation_78116865179938
// MI455X (gfx1250) — compile-verified
//
#include <hip/hip_runtime.h>

// LDPC belief propagation, (7,4) Hamming code, B=262144, 5 iterations.
// One batch per lane; async global<->LDS staging (gfx1250 ASYNCcnt path).

#define NV 7
#define NC 3
#define ITERS 5
#define TPB 256            // threads per block == batches per block
#define CHUNK (TPB * NV)   // floats per block chunk (1792 floats = 7168 B)

#if __has_builtin(__builtin_amdgcn_global_load_async_to_lds_b32) && \
    __has_builtin(__builtin_amdgcn_global_store_async_from_lds_b32)
#define HAVE_ASYNC_LDS 1
#else
#define HAVE_ASYNC_LDS 0
#pragma message("gfx1250 async-LDS builtins NOT found; using synchronous staging fallback")
#endif

// Builtin param types per hipcc diagnostic: (AS1 int*, AS3 int*, imm, imm)
typedef __attribute__((address_space(1))) int* as1_int_t;
typedef __attribute__((address_space(3))) int* as3_int_t;

static __device__ __forceinline__ void wait_async0() {
#if __has_builtin(__builtin_amdgcn_s_wait_asynccnt)
  __builtin_amdgcn_s_wait_asynccnt(0);
#else
  asm volatile("s_wait_asynccnt 0" ::: "memory");
#endif
}

// gfx1250 has native v_tanh_f32; fall back to exp2-based tanh otherwise.
static __device__ __forceinline__ float fast_tanh(float x) {
#if __has_builtin(__builtin_amdgcn_tanhf)
  return __builtin_amdgcn_tanhf(x);
#elif __has_builtin(__builtin_amdgcn_tanh_f32)
  return __builtin_amdgcn_tanh_f32(x);
#else
  // tanh(x) = 1 - 2/(e^{2x}+1),  e^{2x} = exp2(2*log2(e)*x)
  float e = __builtin_amdgcn_exp2f(2.8853900817779268f * x);
  return 1.0f - 2.0f * __builtin_amdgcn_rcpf(e + 1.0f);
#endif
}

// phi(x) = 2*atanh(tanh(x/2)) = ln((1+y)/(1-y)), y = tanh(x/2)
static __device__ __forceinline__ float phi_fn(float x) {
  float y = fast_tanh(0.5f * x);
  float z = (1.0f + y) * __builtin_amdgcn_rcpf(1.0f - y);
  return 0.69314718055994531f * __builtin_amdgcn_logf(z);  // ln2 * log2(z)
}

// Full dense (3,7) message update, matching the reference semantics
// (including phi(1.0) contributions at unmasked positions).
static __device__ __forceinline__ void bp_compute(const float (&L)[NV], float (&res)[NV]) {
  const bool mask[NC][NV] = {
      {true,  false, true,  false, true,  false, true },
      {false, true,  true,  false, false, true,  true },
      {false, false, false, true,  true,  true,  true }};

  float m[NC][NV];
#pragma unroll
  for (int c = 0; c < NC; ++c)
#pragma unroll
    for (int v = 0; v < NV; ++v) m[c][v] = 0.0f;

  const float one_msg = phi_fn(1.0f);  // value the reference puts at unmasked lanes

#pragma unroll
  for (int it = 0; it < ITERS; ++it) {
    float mnew[NC][NV];
#pragma unroll
    for (int c = 0; c < NC; ++c) {
      float tt[NV];
      float prod = 1.0f;
#pragma unroll
      for (int v = 0; v < NV; ++v) {
        tt[v] = 1.0f;
        if (mask[c][v]) {
          tt[v] = fast_tanh(0.5f * (L[v] + m[c][v]));
          prod *= tt[v];
        }
      }
#pragma unroll
      for (int v = 0; v < NV; ++v) {
        mnew[c][v] = mask[c][v]
                         ? phi_fn(prod * __builtin_amdgcn_rcpf(tt[v] + 1e-10f))
                         : one_msg;
      }
    }
    float tot[NV];
#pragma unroll
    for (int v = 0; v < NV; ++v)
      tot[v] = L[v] + mnew[0][v] + mnew[1][v] + mnew[2][v];
#pragma unroll
    for (int c = 0; c < NC; ++c)
#pragma unroll
      for (int v = 0; v < NV; ++v) m[c][v] = tot[v] - mnew[c][v];
  }
#pragma unroll
  for (int v = 0; v < NV; ++v) res[v] = L[v] + m[0][v] + m[1][v] + m[2][v];
}

__global__ __launch_bounds__(TPB) void ldpc_bp_kernel(const float* __restrict__ llr,
                                                      float* __restrict__ out,
                                                      int nbatch) {
  __shared__ float lds[CHUNK];
  const int t = threadIdx.x;
  const long long chunk = (long long)blockIdx.x * CHUNK;
  const int batches_here = nbatch - blockIdx.x * TPB;

  float L[NV], res[NV];

  if (batches_here >= TPB) {
    // ---- stage 7168 B of llr into LDS, lane-contiguous (coalesced) ----
#if HAVE_ASYNC_LDS
    const float* g = llr + chunk;
#pragma unroll
    for (int k = 0; k < NV; ++k) {
      __builtin_amdgcn_global_load_async_to_lds_b32(
          (as1_int_t)(int*)(g + t + k * TPB),
          (as3_int_t)(int*)&lds[t + k * TPB],
          /*offset=*/0, /*cpol=*/0);
    }
    wait_async0();
    __syncthreads();
#else
#pragma unroll
    for (int k = 0; k < NV; ++k) lds[t + k * TPB] = llr[chunk + t + k * TPB];
    __syncthreads();
#endif

    // stride-7 LDS read: 7 coprime with 64 banks -> conflict-free
#pragma unroll
    for (int v = 0; v < NV; ++v) L[v] = lds[t * NV + v];

    bp_compute(L, res);

    // thread t only overwrites the lds region it read -> no barrier needed here
#pragma unroll
    for (int v = 0; v < NV; ++v) lds[t * NV + v] = res[v];
    __syncthreads();

    // ---- stream results back out, coalesced ----
#if HAVE_ASYNC_LDS
    float* o = out + chunk;
#pragma unroll
    for (int k = 0; k < NV; ++k) {
      __builtin_amdgcn_global_store_async_from_lds_b32(
          (as1_int_t)(int*)(o + t + k * TPB),
          (as3_int_t)(int*)&lds[t + k * TPB],
          /*offset=*/0, /*cpol=*/0);
    }
    wait_async0();
#else
#pragma unroll
    for (int k = 0; k < NV; ++k) out[chunk + t + k * TPB] = lds[t + k * TPB];
#endif
  } else {
    // tail path (not taken for B=262144; kept for safety)
    int b = blockIdx.x * TPB + t;
    if (b < nbatch) {
#pragma unroll
      for (int v = 0; v < NV; ++v) L[v] = llr[(long long)b * NV + v];
      bp_compute(L, res);
#pragma unroll
      for (int v = 0; v < NV; ++v) out[(long long)b * NV + v] = res[v];
    }
  }
}

extern "C" void kernel_launch(void* const* d_in, const int* in_sizes, int n_in,
                              void* d_out, int out_size, void* d_ws, size_t ws_size,
                              hipStream_t stream) {
  const float* llr = (const float*)d_in[0];
  // d_in[1] is max_iters (device scalar); reference fixes it at 5 -> ITERS.
  float* out = (float*)d_out;
  const int nbatch = in_sizes[0] / NV;  // 262144
  const int blocks = (nbatch + TPB - 1) / TPB;
  ldpc_bp_kernel<<<blocks, TPB, 0, stream>>>(llr, out, nbatch);
}